// SSKernelNPLR_9990093930981
// MI455X (gfx1250) — compile-verified
//
#include <hip/hip_runtime.h>
#include <hip/hip_bf16.h>

// Problem dims (fixed by reference): H=512, N=64, R=1, CH=1, L=4096
#define HH   512
#define NN   64
#define LL   4096
#define LF   (LL/2 + 1)     // 2049 rfft bins
#define KPAD 2052           // pad K to multiple of 4 for 16x16x4 WMMA steps
#define STEPS (KPAD/4)      // 513 K-steps
#define RESYNC 128          // phasor resync interval (steps)

typedef float v2f __attribute__((ext_vector_type(2)));
typedef float v8f __attribute__((ext_vector_type(8)));

static __device__ __forceinline__ float2 cmul(float ar, float ai, float br, float bi) {
    return make_float2(ar*br - ai*bi, ar*bi + ai*br);
}

// ---------------------------------------------------------------------------
// Kernel 1: Cauchy sums + Woodbury correction -> weighted spectrum.
// One block per feature h; poles & outer-product terms staged in LDS.
// z_l = 2i*tan(pi*l/L) (purely imaginary), den = 1 + e^{-i*2pi*l/L}.
// Stores  kfRe =  Re(kf)*c_l/L   and   kfImN = -Im(kf)*c_l/L
// (imag sign pre-folded so kernel 2's B-fragment is the raw phasor).
// ---------------------------------------------------------------------------
__global__ __launch_bounds__(256) void cauchy_woodbury_kernel(
    const float* __restrict__ log_dt,
    const float* __restrict__ w_re, const float* __restrict__ w_im,
    const float* __restrict__ B_re, const float* __restrict__ B_im,
    const float* __restrict__ P_re, const float* __restrict__ P_im,
    const float* __restrict__ C_re, const float* __restrict__ C_im,
    float* __restrict__ kfRe, float* __restrict__ kfImN)
{
    __shared__ float s_v[8][NN];   // v00re,v00im,v01re,v01im,v10re,v10im,v11re,v11im
    __shared__ float s_w[2][NN];   // w_re*dt, w_im*dt

    const int h   = blockIdx.x;
    const int tid = threadIdx.x;
    const float dt = expf(log_dt[h]);

    if (tid < NN) {
        const int n = tid;
        const float br = B_re[h*NN + n], bi = B_im[h*NN + n];
        const float pr = P_re[h*NN + n], pi = P_im[h*NN + n];
        const float cr = C_re[h*NN + n], ci = C_im[h*NN + n];
        const float qr = pr, qi = -pi;                  // Q = conj(P)
        float2 t;
        t = cmul(br, bi, cr, ci); s_v[0][n] = t.x; s_v[1][n] = t.y;  // v00 = B*C
        t = cmul(br, bi, qr, qi); s_v[2][n] = t.x; s_v[3][n] = t.y;  // v01 = B*Q
        t = cmul(pr, pi, cr, ci); s_v[4][n] = t.x; s_v[5][n] = t.y;  // v10 = P*C
        t = cmul(pr, pi, qr, qi); s_v[6][n] = t.x; s_v[7][n] = t.y;  // v11 = P*Q
        s_w[0][n] = w_re[h*NN + n] * dt;
        s_w[1][n] = w_im[h*NN + n] * dt;
    }
    __syncthreads();

    const float TWO_PI = 6.28318530717958647692f;

    for (int l = tid; l < KPAD; l += 256) {
        float outRe, outIm;
        if (l >= LF) {                        // K padding: zero
            outRe = 0.0f; outIm = 0.0f;
        } else if (l == LL/2) {               // analytic Nyquist: 0.5*dt*sum(v00)
            float sre = 0.0f, sim = 0.0f;
            #pragma unroll 4
            for (int n = 0; n < NN; ++n) { sre += s_v[0][n]; sim += s_v[1][n]; }
            const float wl = 1.0f / (float)LL;     // c_{L/2} = 1
            outRe = 0.5f * dt * sre * wl;
            outIm = 0.5f * dt * sim * wl;
        } else {
            const float phi = (TWO_PI / (float)LL) * (float)l;
            const float zim = 2.0f * tanf(0.5f * phi);   // z = 2i*tan(phi/2)
            float r00r=0.f,r00i=0.f,r01r=0.f,r01i=0.f;
            float r10r=0.f,r10i=0.f,r11r=0.f,r11i=0.f;
            #pragma unroll 4
            for (int n = 0; n < NN; ++n) {
                const float dre = -s_w[0][n];
                const float dim = zim - s_w[1][n];
                const float inv = 1.0f / (dre*dre + dim*dim);
                const float rr =  dre * inv;             // recip = 1/(z - w*dt)
                const float ri = -dim * inv;
                r00r += s_v[0][n]*rr - s_v[1][n]*ri;  r00i += s_v[0][n]*ri + s_v[1][n]*rr;
                r01r += s_v[2][n]*rr - s_v[3][n]*ri;  r01i += s_v[2][n]*ri + s_v[3][n]*rr;
                r10r += s_v[4][n]*rr - s_v[5][n]*ri;  r10i += s_v[4][n]*ri + s_v[5][n]*rr;
                r11r += s_v[6][n]*rr - s_v[7][n]*ri;  r11i += s_v[6][n]*ri + s_v[7][n]*rr;
            }
            r00r*=dt; r00i*=dt; r01r*=dt; r01i*=dt;
            r10r*=dt; r10i*=dt; r11r*=dt; r11i*=dt;
            // Woodbury: kf = r00 - r01*r10/(1 + r11)
            const float2 tnum = cmul(r01r, r01i, r10r, r10i);
            const float ur = 1.0f + r11r, ui = r11i;
            const float uinv = 1.0f / (ur*ur + ui*ui);
            const float dr = (tnum.x*ur + tnum.y*ui) * uinv;
            const float di = (tnum.y*ur - tnum.x*ui) * uinv;
            float kfr = r00r - dr, kfi = r00i - di;
            // kf *= 2/den, den = 1 + e^{-i*phi} = (1+cos) - i*sin
            float sphi, cphi;
            __sincosf(phi, &sphi, &cphi);
            const float denr = 1.0f + cphi;
            const float dinv = 2.0f / (denr*denr + sphi*sphi);
            const float mr = denr * dinv;     // 2*conj(den)/|den|^2
            const float mi = sphi * dinv;
            outRe = kfr*mr - kfi*mi;
            outIm = kfr*mi + kfi*mr;
            const float wl = (l == 0) ? (1.0f/(float)LL) : (2.0f/(float)LL);
            outRe *= wl; outIm *= wl;
        }
        kfRe [h*KPAD + l] =  outRe;
        kfImN[h*KPAD + l] = -outIm;          // sign folded for kernel 2
    }
}

// ---------------------------------------------------------------------------
// Kernel 2: inverse real DFT as fused f32 WMMA GEMMs.
// k[h,t] = sum_l kfRe[h,l]*cos(2pi*l*t/L) + kfImN[h,l]*sin(2pi*l*t/L)
// One wave per 16x16 (h,t) tile; K = 2052 consumed in 16x16x4 steps.
// Basis generated in-register via phasor rotation; resync every 128 steps
// using EXACT integer angle reduction: theta*l mod 2pi = (2pi/L)*((t*l)&(L-1)).
// ---------------------------------------------------------------------------
__global__ __launch_bounds__(256) void irdft_wmma_kernel(
    const float* __restrict__ kfRe, const float* __restrict__ kfImN,
    float* __restrict__ out)
{
    const int wave = threadIdx.x >> 5;
    const int lane = threadIdx.x & 31;
    const int tile = blockIdx.x * 8 + wave;     // 8192 tiles total
    const int hb   = tile >> 8;                 // 256 t-tiles per h-block row
    const int tb   = tile & 255;
    const int hbase = hb * 16;
    const int tbase = tb * 16;

    const int m    = lane & 15;                 // A row / B col / D col index
    const int half = lane >> 4;                 // K-pair select per ISA layout
    const int d0 = 2*half, d1 = 2*half + 1;     // absolute K offsets this lane holds

    // A fragment for a lane is exactly elements (l0+2*half, l0+2*half+1):
    // load them directly as float2 -> global_load_b64, no selects.
    const float2* __restrict__ rowRe = (const float2*)(kfRe  + (size_t)(hbase + m) * KPAD);
    const float2* __restrict__ rowIm = (const float2*)(kfImN + (size_t)(hbase + m) * KPAD);

    const int t = tbase + m;
    const float STEP_ANG = 6.28318530717958647692f / (float)LL;   // 2pi/L

    float c4, s4;                               // rotation by 4 bins: e^{i*4*theta}
    __sincosf(STEP_ANG * (float)((4 * t) & (LL - 1)), &s4, &c4);

    v8f acc = {};

    for (int blk = 0; blk < STEPS; blk += RESYNC) {
        // exact integer-reduced resync: angle = (2pi/L) * ((t*l) mod L)
        float p0r, p0i, p1r, p1i;
        __sincosf(STEP_ANG * (float)((t * (4*blk + d0)) & (LL - 1)), &p0i, &p0r);
        __sincosf(STEP_ANG * (float)((t * (4*blk + d1)) & (LL - 1)), &p1i, &p1r);

        const int end = (blk + RESYNC < STEPS) ? (blk + RESYNC) : STEPS;
        #pragma unroll 4
        for (int s = blk; s < end; ++s) {
            const int l0 = s << 2;
            v2f Ar; { float2 a = rowRe[(l0 >> 1) + half]; Ar[0] = a.x; Ar[1] = a.y; }
            v2f Ai; { float2 a = rowIm[(l0 >> 1) + half]; Ai[0] = a.x; Ai[1] = a.y; }
            v2f Bc; Bc[0] = p0r; Bc[1] = p1r;   // cos row pair
            v2f Bs; Bs[0] = p0i; Bs[1] = p1i;   // sin row pair (sign pre-folded)

            acc = __builtin_amdgcn_wmma_f32_16x16x4_f32(false, Ar, false, Bc,
                                                        (short)0, acc, false, false);
            acc = __builtin_amdgcn_wmma_f32_16x16x4_f32(false, Ai, false, Bs,
                                                        (short)0, acc, false, false);
            // advance phasors by 4 frequency bins
            const float n0r = p0r*c4 - p0i*s4, n0i = p0r*s4 + p0i*c4;
            const float n1r = p1r*c4 - p1i*s4, n1i = p1r*s4 + p1i*c4;
            p0r = n0r; p0i = n0i; p1r = n1r; p1i = n1i;
        }
    }

    // D 16x16 layout: VGPR j -> row (j + 8*half), col = lane&15
    #pragma unroll
    for (int j = 0; j < 8; ++j) {
        out[(size_t)(hbase + j + 8*half) * LL + (size_t)(tbase + m)] = acc[j];
    }
}

extern "C" void kernel_launch(void* const* d_in, const int* in_sizes, int n_in,
                              void* d_out, int out_size, void* d_ws, size_t ws_size,
                              hipStream_t stream) {
    const float* log_dt = (const float*)d_in[0];
    const float* w_re   = (const float*)d_in[1];
    const float* w_im   = (const float*)d_in[2];
    const float* B_re   = (const float*)d_in[3];
    const float* B_im   = (const float*)d_in[4];
    const float* P_re   = (const float*)d_in[5];
    const float* P_im   = (const float*)d_in[6];
    const float* C_re   = (const float*)d_in[7];
    const float* C_im   = (const float*)d_in[8];
    // d_in[9] = L scalar (4096, baked in as LL)

    float* kfRe  = (float*)d_ws;                      // H*KPAD floats (~4.2 MB)
    float* kfImN = kfRe + (size_t)HH * KPAD;          // H*KPAD floats (~4.2 MB)
    float* out   = (float*)d_out;                     // CH*H*L = 512*4096 floats

    cauchy_woodbury_kernel<<<HH, 256, 0, stream>>>(
        log_dt, w_re, w_im, B_re, B_im, P_re, P_im, C_re, C_im, kfRe, kfImN);

    // 8192 tiles (32 h-blocks x 256 t-blocks), 8 waves per block
    irdft_wmma_kernel<<<(HH/16) * (LL/16) / 8, 256, 0, stream>>>(kfRe, kfImN, out);
}